// GCN_63513976373672
// MI455X (gfx1250) — compile-verified
//
#include <hip/hip_runtime.h>

// CDNA5 / gfx1250 GCN pipeline.
// GEMMs use V_WMMA_F32_16X16X4_F32 (f32 in, f32 accum) so results match the
// f32 JAX reference; graph aggregation uses global f32 atomics.

typedef __attribute__((ext_vector_type(2))) float v2f;
typedef __attribute__((ext_vector_type(8))) float v8f;

#define N_USER 100000
#define N_ITEM 100000
#define NNODES (N_USER + N_ITEM)
#define HID    128
#define E_UI   600000
#define E_IU   600000
#define E_TOT  (E_UI + E_IU)

// ---------------------------------------------------------------------------
// GEMM: C[M x 128] = (relu? max(A,0):A)[M x K] @ W[K x 128] (+ bias[128])
// 128 threads = 4 waves; wave w computes rows [blk*64 + 16w, +16), all 128 cols.
// K must be a multiple of 4. EXEC stays all-ones through the WMMA loop
// (row indices are clamped for tail blocks; stores are predicated afterwards).
// ---------------------------------------------------------------------------
__global__ __launch_bounds__(128) void gemm_n128(
    const float* __restrict__ A, const float* __restrict__ W,
    const float* __restrict__ bias, float* __restrict__ C,
    int M, int K, int relu, int use_bias)
{
  const int lane  = threadIdx.x & 31;
  const int wave  = threadIdx.x >> 5;          // 0..3
  const long row0 = (long)blockIdx.x * 64 + (long)wave * 16;

  const int m     = lane & 15;                 // A row within tile
  const int n     = lane & 15;                 // B/C col within tile
  const int khalf = (lane >> 4) << 1;          // 0 or 2 (A/B K sub-lane)

  long arow = row0 + m;
  if (arow >= M) arow = M - 1;                 // clamp: keep loads in-bounds, EXEC full
  const float* __restrict__ Ar = A + arow * (long)K;

  v8f acc[8] = {};                             // 16 rows x 128 cols per wave

  for (int k = 0; k < K; k += 4) {
    // A fragment 16x4: lanes 0-15 -> K k+0,k+1 ; lanes 16-31 -> K k+2,k+3
    v2f a;
    a.x = Ar[k + khalf];
    a.y = Ar[k + khalf + 1];
    if (relu) { a.x = fmaxf(a.x, 0.0f); a.y = fmaxf(a.y, 0.0f); }

    const float* __restrict__ Wk = W + (long)(k + khalf) * HID + n;
#pragma unroll
    for (int t = 0; t < 8; ++t) {
      // B fragment 4x16 of column tile t: VGPR0 = row k+khalf, VGPR1 = row k+khalf+1
      v2f b;
      b.x = Wk[t * 16];
      b.y = Wk[t * 16 + HID];
      acc[t] = __builtin_amdgcn_wmma_f32_16x16x4_f32(
          /*neg_a=*/false, a, /*neg_b=*/false, b,
          /*c_mod=*/(short)0, acc[t], /*reuse_a=*/false, /*reuse_b=*/false);
    }
  }

  // C/D layout: VGPR r = row (r) for lanes 0-15, row (r+8) for lanes 16-31
  const int rhalf = (lane >> 4) << 3;          // 0 or 8
#pragma unroll
  for (int t = 0; t < 8; ++t) {
    float bv = use_bias ? bias[t * 16 + n] : 0.0f;
#pragma unroll
    for (int r = 0; r < 8; ++r) {
      long row = row0 + rhalf + r;
      if (row < M) C[row * HID + t * 16 + n] = acc[t][r] + bv;
    }
  }
}

// ---------------------------------------------------------------------------
// Degree / normalization
// ---------------------------------------------------------------------------
__global__ void deg_init(float* __restrict__ deg, int n)
{
  int i = blockIdx.x * blockDim.x + threadIdx.x;
  if (i < n) deg[i] = 1.0f;                    // self loop contribution
}

__global__ void deg_count(const int* __restrict__ eui,
                          const int* __restrict__ eiu,
                          float* __restrict__ deg)
{
  int e = blockIdx.x * blockDim.x + threadIdx.x;
  if (e >= E_TOT) return;
  int d;
  if (e < E_UI) d = eui[E_UI + e] + N_USER;    // edge_ui[1][e] + N_USER
  else          d = eiu[E_IU + (e - E_UI)];    // edge_iu[1][e']
  atomicAdd(&deg[d], 1.0f);
}

__global__ void dinv_fin(float* __restrict__ deg, int n)
{
  int i = blockIdx.x * blockDim.x + threadIdx.x;
  if (i < n) {
    float v = deg[i];
    deg[i] = v > 0.0f ? rsqrtf(v) : 0.0f;
  }
}

// ---------------------------------------------------------------------------
// Aggregation: out = bias + dinv[i]^2 * hW[i]  (self loop + bias init)
// ---------------------------------------------------------------------------
__global__ void agg_self(const float* __restrict__ hw,
                         const float* __restrict__ dinv,
                         const float* __restrict__ bias,
                         float* __restrict__ out)
{
  long idx = (long)blockIdx.x * blockDim.x + threadIdx.x;
  if (idx >= (long)NNODES * HID) return;
  long i = idx >> 7;
  int  c = (int)(idx & 127);
  float di = dinv[i];
  out[idx] = bias[c] + di * di * hw[idx];
}

// ---------------------------------------------------------------------------
// Edge scatter: out[d] += dinv[s]*dinv[d] * hW[s]
// One wave per edge: each lane loads float4 (512B/wave gather) and does 4
// global f32 atomic adds. 8 edges per 256-thread block.
// ---------------------------------------------------------------------------
__global__ __launch_bounds__(256) void agg_edges(
    const int* __restrict__ eui, const int* __restrict__ eiu,
    const float* __restrict__ hw, const float* __restrict__ dinv,
    float* __restrict__ out)
{
  int lane = threadIdx.x & 31;
  int w    = threadIdx.x >> 5;
  int e    = blockIdx.x * 8 + w;
  if (e >= E_TOT) return;

  int s, d;
  if (e < E_UI) { s = eui[e];                 d = eui[E_UI + e] + N_USER; }
  else          { int e2 = e - E_UI;
                  s = eiu[e2] + N_USER;       d = eiu[E_IU + e2]; }

  float nrm = dinv[s] * dinv[d];
  const float4* __restrict__ hs = (const float4*)(hw + (long)s * HID);
  float4 v = hs[lane];
  float* od = out + (long)d * HID + lane * 4;
  atomicAdd(od + 0, nrm * v.x);
  atomicAdd(od + 1, nrm * v.y);
  atomicAdd(od + 2, nrm * v.z);
  atomicAdd(od + 3, nrm * v.w);
}

// ---------------------------------------------------------------------------
// Host launch
// ---------------------------------------------------------------------------
extern "C" void kernel_launch(void* const* d_in, const int* in_sizes, int n_in,
                              void* d_out, int out_size, void* d_ws, size_t ws_size,
                              hipStream_t stream)
{
  const float* x_user = (const float*)d_in[0];
  const float* x_item = (const float*)d_in[1];
  const int*   eui    = (const int*)d_in[2];
  const int*   eiu    = (const int*)d_in[3];
  const float* W_iu   = (const float*)d_in[4];
  const float* b_iu   = (const float*)d_in[5];
  const float* W_ii   = (const float*)d_in[6];
  const float* b_ii   = (const float*)d_in[7];
  const float* W1     = (const float*)d_in[8];
  const float* b1     = (const float*)d_in[9];
  const float* W2     = (const float*)d_in[10];
  const float* b2     = (const float*)d_in[11];
  float* out = (float*)d_out;

  // Workspace: h (NNODES*HID) | hw (NNODES*HID) | dinv (NNODES)  ~= 206 MB
  float* h    = (float*)d_ws;
  float* hw   = h  + (size_t)NNODES * HID;
  float* dinv = hw + (size_t)NNODES * HID;

  const dim3 blk(128);

  // 1) Per-type input projections (with bias): h = x @ W_in + b_in
  gemm_n128<<<dim3((N_USER + 63) / 64), blk, 0, stream>>>(
      x_user, W_iu, b_iu, h, N_USER, 256, /*relu=*/0, /*use_bias=*/1);
  gemm_n128<<<dim3((N_ITEM + 63) / 64), blk, 0, stream>>>(
      x_item, W_ii, b_ii, h + (size_t)N_USER * HID, N_ITEM, 256, 0, 1);

  // 2) Symmetric normalization from dst degrees (+ self loops)
  deg_init <<<dim3((NNODES + 255) / 256), dim3(256), 0, stream>>>(dinv, NNODES);
  deg_count<<<dim3((E_TOT + 255) / 256), dim3(256), 0, stream>>>(eui, eiu, dinv);
  dinv_fin <<<dim3((NNODES + 255) / 256), dim3(256), 0, stream>>>(dinv, NNODES);

  const unsigned aggBlocks  = (unsigned)(((size_t)NNODES * HID + 255) / 256);
  const unsigned edgeBlocks = (E_TOT + 7) / 8;

  // 3) Layer 1: hw = h @ W1 ; h = b1 + dinv^2*hw ; scatter edges into h
  gemm_n128<<<dim3((NNODES + 63) / 64), blk, 0, stream>>>(
      h, W1, b1, hw, NNODES, HID, /*relu=*/0, /*use_bias=*/0);
  agg_self <<<dim3(aggBlocks),  dim3(256), 0, stream>>>(hw, dinv, b1, h);
  agg_edges<<<dim3(edgeBlocks), dim3(256), 0, stream>>>(eui, eiu, hw, dinv, h);

  // 4) Layer 2: hw = relu(h) @ W2 ; out = b2 + dinv^2*hw ; scatter edges
  gemm_n128<<<dim3((NNODES + 63) / 64), blk, 0, stream>>>(
      h, W2, b2, hw, NNODES, HID, /*relu=*/1, /*use_bias=*/0);
  agg_self <<<dim3(aggBlocks),  dim3(256), 0, stream>>>(hw, dinv, b2, out);
  agg_edges<<<dim3(edgeBlocks), dim3(256), 0, stream>>>(eui, eiu, hw, dinv, out);
}